// ComputeLoss3d_12506944766654
// MI455X (gfx1250) — compile-verified
//
#include <hip/hip_runtime.h>

typedef __attribute__((ext_vector_type(2))) float v2f;
typedef __attribute__((ext_vector_type(8))) float v8f;

#define WAVES_PER_BLOCK 8
#define BLOCK_THREADS (WAVES_PER_BLOCK * 32)
#define COL_CHUNK 2048                 // columns staged per LDS pass (32 KB)

// Block = 8 waves = 128 rows of P1 (16 rows per wave).  The block stages
// COL_CHUNK columns of P2 into LDS as (x,y,z,|q|^2); each wave sweeps them
// with V_WMMA_F32_16X16X4_F32 where the K=3 slot is repurposed so the WMMA
// computes the full expression directly:
//   A = {-2x, -2y, -2z, 1},  B = {x', y', z', |q|^2}
//   =>  C[m,n] = |q_n|^2 - 2<p_m, q_n>
// and the row-min only needs min ops; |p|^2 is added once after the sweep:
//   min_n dist = |p|^2 + min_n C[m,n]
__global__ __launch_bounds__(BLOCK_THREADS)
void chamfer_rowmin_wmma(const float* __restrict__ P1,   // [B, Nrows, 3]
                         const float* __restrict__ P2,   // [B, Mcols, 3]
                         int Nrows, int Mcols,
                         float* __restrict__ partials,   // one float per wave
                         int partialBase)
{
    __shared__ float4 scol[COL_CHUNK];                   // (x, y, z, |q|^2)

    const int tid  = threadIdx.x;
    const int lane = tid & 31;
    const int wave = tid >> 5;

    const int rowGroups = Nrows >> 7;                    // 128 rows / block
    const int b  = blockIdx.x / rowGroups;
    const int rg = blockIdx.x % rowGroups;
    const int rowBase = ((rg << 3) + wave) << 4;         // this wave's 16 rows

    const float* p1 = P1 + (size_t)b * Nrows * 3;
    const float* p2 = P2 + (size_t)b * Mcols * 3;

    // ---- A fragment (16x4 f32, ISA layout): lanes 0-15 -> M=lane, K={0,1};
    //      lanes 16-31 -> M=lane-16, K={2,3}.  Scaled by -2, K=3 slot = 1 so
    //      B's K=3 slot (|q|^2) is accumulated with weight 1.
    const bool hiK = lane >= 16;
    const float* arow = p1 + (size_t)(rowBase + (lane & 15)) * 3;
    v2f afrag;
    afrag.x = hiK ? (-2.0f * arow[2]) : (-2.0f * arow[0]);
    afrag.y = hiK ? 1.0f              : (-2.0f * arow[1]);

    // per-lane LDS byte base: column (lane%16); lo lanes read {x,y} at +0,
    // hi lanes read {z, |q|^2} at +8  ->  B = {x',y',z',|q|^2} per ISA layout
    const char* sbase = reinterpret_cast<const char*>(scol)
                      + ((lane & 15) << 4) + (hiK ? 8 : 0);

    float rmin[8];
#pragma unroll
    for (int v = 0; v < 8; ++v) rmin[v] = 3.0e38f;

    for (int c0 = 0; c0 < Mcols; c0 += COL_CHUNK) {
        __syncthreads();                                 // WAR vs prev chunk
        for (int i = tid; i < COL_CHUNK; i += BLOCK_THREADS) {
            const float* q = p2 + (size_t)(c0 + i) * 3;
            const float x = q[0], y = q[1], z = q[2];
            scol[i] = make_float4(x, y, z, x * x + y * y + z * z);
        }
        __syncthreads();

        const int nColTiles = COL_CHUNK >> 4;            // 16 cols = 256 B
#pragma unroll 4
        for (int t = 0; t < nColTiles; ++t) {
            const v2f bfrag = *reinterpret_cast<const v2f*>(sbase + (t << 8));

            v8f c = {};
            // (neg_a, A, neg_b, B, c_mod, C, reuse_a, reuse_b)
            c = __builtin_amdgcn_wmma_f32_16x16x4_f32(
                    false, afrag, false, bfrag, (short)0, c, false, false);

            // C layout: lane l, VGPR v -> row = rowBase + v + 8*(l/16)
#pragma unroll
            for (int v = 0; v < 8; ++v)
                rmin[v] = fminf(rmin[v], c[v]);
        }
    }

    // min across the 16 lanes that share each row (xor stays within a half)
#pragma unroll
    for (int v = 0; v < 8; ++v) {
        float r = rmin[v];
        r = fminf(r, __shfl_xor(r, 1, 32));
        r = fminf(r, __shfl_xor(r, 2, 32));
        r = fminf(r, __shfl_xor(r, 4, 32));
        r = fminf(r, __shfl_xor(r, 8, 32));
        rmin[v] = r;
    }

    // each half-wave sums its 8 rows (adding |p1|^2), then combine halves
    const int rhalf = rowBase + ((lane >> 4) << 3);      // rows rhalf..rhalf+7
    float s = 0.0f;
#pragma unroll
    for (int v = 0; v < 8; ++v) {
        const float* pr = p1 + (size_t)(rhalf + v) * 3;
        const float sq1 = pr[0] * pr[0] + pr[1] * pr[1] + pr[2] * pr[2];
        s += rmin[v] + sq1;
    }
    s += __shfl_xor(s, 16, 32);                          // halves: disjoint rows

    if (lane == 0)
        partials[partialBase + (blockIdx.x * WAVES_PER_BLOCK + wave)] = s;
}

// Deterministic fixed-order reduction of all per-wave partials.
__global__ __launch_bounds__(256)
void chamfer_finalize(const float* __restrict__ partials, int n,
                      float* __restrict__ out)
{
    __shared__ float red[256];
    float s = 0.0f;
    for (int i = threadIdx.x; i < n; i += 256) s += partials[i];
    red[threadIdx.x] = s;
    __syncthreads();
    for (int off = 128; off > 0; off >>= 1) {
        if (threadIdx.x < off) red[threadIdx.x] += red[threadIdx.x + off];
        __syncthreads();
    }
    if (threadIdx.x == 0) out[0] = red[0] * 0.125f;      // divide by B = 8
}

extern "C" void kernel_launch(void* const* d_in, const int* in_sizes, int n_in,
                              void* d_out, int out_size, void* d_ws, size_t ws_size,
                              hipStream_t stream) {
    const float* gt = (const float*)d_in[0];   // [B, N, 3]
    const float* sp = (const float*)d_in[1];   // [B, M, 3]
    float* out      = (float*)d_out;
    float* partials = (float*)d_ws;

    const int B = 8;
    const int N = in_sizes[0] / (B * 3);       // 8192
    const int M = in_sizes[1] / (B * 3);       // 2048

    const int waves1 = B * (N >> 4);           // term 1: rows from gt
    const int waves2 = B * (M >> 4);           // term 2: rows from struct

    chamfer_rowmin_wmma<<<waves1 / WAVES_PER_BLOCK, BLOCK_THREADS, 0, stream>>>(
        gt, sp, N, M, partials, 0);
    chamfer_rowmin_wmma<<<waves2 / WAVES_PER_BLOCK, BLOCK_THREADS, 0, stream>>>(
        sp, gt, M, N, partials, waves1);

    chamfer_finalize<<<1, 256, 0, stream>>>(partials, waves1 + waves2, out);
}